// MultiHeadCrossAttention_75969381531890
// MI455X (gfx1250) — compile-verified
//
#include <hip/hip_runtime.h>
#include <hip/hip_bf16.h>
#include <math.h>

typedef __attribute__((ext_vector_type(2))) float        v2f;
typedef __attribute__((ext_vector_type(8))) float        v8f;
typedef __attribute__((ext_vector_type(4))) unsigned int v4u;
typedef __attribute__((ext_vector_type(4))) int          v4i;
typedef __attribute__((ext_vector_type(8))) int          v8i;

#define H_   1024
#define NH_  16
#define HD_  64
#define B_   2
#define LQ_  1024
#define LK_  2048

#if defined(__has_builtin)
#if __has_builtin(__builtin_amdgcn_tensor_load_to_lds)
#define HAVE_TDM 1
#endif
#endif

// D = A(16x4) * B(4x16) + C, fp32 matrix pipe.
__device__ __forceinline__ v8f wmma_f32(v2f a, v2f b, v8f c) {
    return __builtin_amdgcn_wmma_f32_16x16x4_f32(
        /*neg_a=*/false, a, /*neg_b=*/false, b,
        /*c_mod=*/(short)0, c, /*reuse_a=*/false, /*reuse_b=*/false);
}

#if HAVE_TDM
// Issue a TDM load of a 64-row x 32-float 2D tile (row stride `stride` floats)
// into LDS at byte offset lds_byte_off, padded +2 dwords per 32-dword row
// (LDS row stride becomes 34 floats -> bank spread + 8B alignment preserved).
__device__ __forceinline__ void tdm_load_tile(const float* gptr, unsigned lds_byte_off,
                                              unsigned td0, unsigned td1, unsigned stride) {
    const unsigned long long ga = (unsigned long long)(size_t)gptr;
    v4u g0;
    g0.x = 1u;                                                  // count=1, no gather
    g0.y = lds_byte_off;                                        // lds_addr (bytes)
    g0.z = (unsigned)ga;                                        // global_addr[31:0]
    g0.w = (unsigned)((ga >> 32) & 0x1FFFFFFull) | (2u << 30);  // addr[56:32] | type=2
    v8i g1;
    g1[0] = (int)((2u << 16)    // data_size = 4 bytes
                | (1u << 20)    // pad_enable
                | (4u << 22)    // pad_interval: every 32 dwords
                | (1u << 25));  // pad_amount: 2 dwords
    g1[1] = (int)(td0 << 16);                               // tensor_dim0[15:0]
    g1[2] = (int)((td0 >> 16) | ((td1 & 0xffffu) << 16));   // td0 hi | tensor_dim1 lo
    g1[3] = (int)((td1 >> 16) | (32u << 16));               // td1 hi | tile_dim0 = 32
    g1[4] = 64;                                             // tile_dim1 = 64 rows
    g1[5] = (int)stride;                                    // tensor_dim0_stride lo
    g1[6] = 0;
    g1[7] = 0;
    v4i z4 = {0, 0, 0, 0};
    v8i z8 = {0, 0, 0, 0, 0, 0, 0, 0};
    __builtin_amdgcn_tensor_load_to_lds(g0, g1, z4, z4, z8, 0);
}
#endif

#define KT          32
#define TILE_PAD    34                 // 32 data + 2 pad dwords per row
#define TILE_FLOATS (64 * TILE_PAD)    // 2176 floats per staged tile

// out[m,n] = sum_k X[m,k] * W[n,k] + bias[n] (+ R[m,n] if R != nullptr)
// Block: 128 threads = 4 waves, 64x64 block tile, 32x32 wave tile (2x2 WMMA tiles).
// A/B tiles staged through LDS by the Tensor Data Mover, double-buffered.
__global__ __launch_bounds__(128)
void gemm_nt_bias(const float* __restrict__ X, const float* __restrict__ W,
                  const float* __restrict__ bias, const float* __restrict__ R,
                  float* __restrict__ out, int M, int N, int K)
{
    const int wave = threadIdx.x >> 5;
    const int lane = threadIdx.x & 31;
    const int lr   = lane & 15;
    const int half = lane >> 4;
    const int m0   = blockIdx.x * 64;
    const int n0   = blockIdx.y * 64;
    const int wm   = (wave >> 1) * 32;
    const int wn   = (wave & 1) * 32;

    v8f acc[2][2];
#pragma unroll
    for (int i = 0; i < 2; ++i)
#pragma unroll
        for (int j = 0; j < 2; ++j)
            acc[i][j] = (v8f)0.0f;

#if HAVE_TDM
    __shared__ float lds[4 * TILE_FLOATS];    // [buf0.A | buf0.B | buf1.A | buf1.B]
    {   // escape LDS: TDM writes are invisible to alias analysis
        void* lp = (void*)lds;
        asm volatile("" : "+v"(lp) :: "memory");
    }
    const float* __restrict__ Arow = X + (size_t)m0 * K;
    const float* __restrict__ Brow = W + (size_t)n0 * K;
    const int NC = K / KT;
    if (wave == 0) {
        tdm_load_tile(Arow, 0u,                       (unsigned)K, (unsigned)M, (unsigned)K);
        tdm_load_tile(Brow, TILE_FLOATS * 4u,         (unsigned)K, (unsigned)N, (unsigned)K);
    }
    for (int c = 0; c < NC; ++c) {
        const int buf = c & 1;
        if (wave == 0) {
            if (c + 1 < NC) {
                const int nb = buf ^ 1;
                tdm_load_tile(Arow + (c + 1) * KT, (unsigned)(2 * nb) * TILE_FLOATS * 4u,
                              (unsigned)K, (unsigned)M, (unsigned)K);
                tdm_load_tile(Brow + (c + 1) * KT, (unsigned)(2 * nb + 1) * TILE_FLOATS * 4u,
                              (unsigned)K, (unsigned)N, (unsigned)K);
                __builtin_amdgcn_s_wait_tensorcnt(2);   // chunk c complete (in-order)
            } else {
                __builtin_amdgcn_s_wait_tensorcnt(0);
            }
        }
        __syncthreads();
        const float* As = &lds[(2 * buf) * TILE_FLOATS];
        const float* Bs = &lds[(2 * buf + 1) * TILE_FLOATS];
#pragma unroll
        for (int ks = 0; ks < KT / 4; ++ks) {
            const int col = ks * 4 + half * 2;              // frag K = 2*half + j
            v2f a0 = *(const v2f*)(As + (wm + lr) * TILE_PAD + col);
            v2f a1 = *(const v2f*)(As + (wm + 16 + lr) * TILE_PAD + col);
            v2f b0 = *(const v2f*)(Bs + (wn + lr) * TILE_PAD + col);
            v2f b1 = *(const v2f*)(Bs + (wn + 16 + lr) * TILE_PAD + col);
            acc[0][0] = wmma_f32(a0, b0, acc[0][0]);
            acc[0][1] = wmma_f32(a0, b1, acc[0][1]);
            acc[1][0] = wmma_f32(a1, b0, acc[1][0]);
            acc[1][1] = wmma_f32(a1, b1, acc[1][1]);
        }
        __syncthreads();
    }
#else
    const float* __restrict__ xr0 = X + (size_t)(m0 + wm + lr) * K;
    const float* __restrict__ xr1 = X + (size_t)(m0 + wm + 16 + lr) * K;
    const float* __restrict__ wr0 = W + (size_t)(n0 + wn + lr) * K;
    const float* __restrict__ wr1 = W + (size_t)(n0 + wn + 16 + lr) * K;
    for (int k0 = 0; k0 < K; k0 += 4) {
        const int kk = k0 + half * 2;
        v2f a0 = *(const v2f*)(xr0 + kk);
        v2f a1 = *(const v2f*)(xr1 + kk);
        v2f b0 = *(const v2f*)(wr0 + kk);
        v2f b1 = *(const v2f*)(wr1 + kk);
        acc[0][0] = wmma_f32(a0, b0, acc[0][0]);
        acc[0][1] = wmma_f32(a0, b1, acc[0][1]);
        acc[1][0] = wmma_f32(a1, b0, acc[1][0]);
        acc[1][1] = wmma_f32(a1, b1, acc[1][1]);
    }
#endif

#pragma unroll
    for (int mt = 0; mt < 2; ++mt)
#pragma unroll
        for (int nt = 0; nt < 2; ++nt)
#pragma unroll
            for (int r = 0; r < 8; ++r) {
                const int m = m0 + wm + mt * 16 + half * 8 + r;  // C: VGPR r -> M=r+8*half
                const int n = n0 + wn + nt * 16 + lr;            //    N = lane%16
                float v = acc[mt][nt][r] + bias[n];
                if (R) v += R[(size_t)m * N + n];
                out[(size_t)m * N + n] = v;
            }
}

// Flash-style attention for one (batch, head, 32-query tile) per single-wave block.
__global__ __launch_bounds__(32)
void attn_head(const float* __restrict__ Qp, const float* __restrict__ Kp,
               const float* __restrict__ Vp, const int* __restrict__ mask,
               float* __restrict__ Ctx)
{
    const int bid  = blockIdx.x;
    const int qt   = bid & 31;
    const int h    = (bid >> 5) & 15;
    const int b    = bid >> 9;
    const int q0   = qt * 32;
    const int lane = threadIdx.x;
    const int lr   = lane & 15;
    const int half = lane >> 4;

    const float* __restrict__ Qb = Qp + (size_t)b * LQ_ * H_ + (size_t)h * HD_;
    const float* __restrict__ Kb = Kp + (size_t)b * LK_ * H_ + (size_t)h * HD_;
    const float* __restrict__ Vb = Vp + (size_t)b * LK_ * H_ + (size_t)h * HD_;
    const int*   __restrict__ Mb = mask + (size_t)b * LQ_ * LK_;

    v2f qf[2][16];
#pragma unroll
    for (int mt = 0; mt < 2; ++mt)
#pragma unroll
        for (int ks = 0; ks < 16; ++ks)
            qf[mt][ks] = *(const v2f*)(Qb + (size_t)(q0 + mt * 16 + lr) * H_ + ks * 4 + half * 2);

    v8f ctx[2][4];
#pragma unroll
    for (int mt = 0; mt < 2; ++mt)
#pragma unroll
        for (int nt = 0; nt < 4; ++nt)
            ctx[mt][nt] = (v8f)0.0f;

    float mrow[2][8], lrow[2][8];
#pragma unroll
    for (int mt = 0; mt < 2; ++mt)
#pragma unroll
        for (int r = 0; r < 8; ++r) { mrow[mt][r] = -1e30f; lrow[mt][r] = 0.0f; }

    __shared__ float ldsP[32 * 16];

    for (int kb = 0; kb < LK_; kb += 16) {
        v8f s[2];
        s[0] = (v8f)0.0f; s[1] = (v8f)0.0f;
#pragma unroll
        for (int ks = 0; ks < 16; ++ks) {
            v2f kf = *(const v2f*)(Kb + (size_t)(kb + lr) * H_ + ks * 4 + half * 2);
            s[0] = wmma_f32(qf[0][ks], kf, s[0]);
            s[1] = wmma_f32(qf[1][ks], kf, s[1]);
        }

#pragma unroll
        for (int mt = 0; mt < 2; ++mt) {
            float alpha[8];
#pragma unroll
            for (int r = 0; r < 8; ++r) {
                const int q = q0 + mt * 16 + half * 8 + r;
                float sv = s[mt][r] * 0.125f;
                const int mv = Mb[(size_t)q * LK_ + kb + lr];
                sv = (mv == 0) ? -1e9f : sv;
                float rm = sv;
                rm = fmaxf(rm, __shfl_xor(rm, 1));
                rm = fmaxf(rm, __shfl_xor(rm, 2));
                rm = fmaxf(rm, __shfl_xor(rm, 4));
                rm = fmaxf(rm, __shfl_xor(rm, 8));
                const float mnew = fmaxf(mrow[mt][r], rm);
                const float p = __expf(sv - mnew);
                float rs = p;
                rs += __shfl_xor(rs, 1);
                rs += __shfl_xor(rs, 2);
                rs += __shfl_xor(rs, 4);
                rs += __shfl_xor(rs, 8);
                const float a = __expf(mrow[mt][r] - mnew);
                lrow[mt][r] = lrow[mt][r] * a + rs;
                mrow[mt][r] = mnew;
                alpha[r] = a;
                s[mt][r] = p;
            }
#pragma unroll
            for (int nt = 0; nt < 4; ++nt)
#pragma unroll
                for (int r = 0; r < 8; ++r)
                    ctx[mt][nt][r] *= alpha[r];
#pragma unroll
            for (int r = 0; r < 8; ++r)
                ldsP[(mt * 16 + half * 8 + r) * 16 + lr] = s[mt][r];
        }
        __syncthreads();

#pragma unroll
        for (int kk = 0; kk < 4; ++kk) {
            v2f pf[2];
#pragma unroll
            for (int mt = 0; mt < 2; ++mt) {
                const int col = kk * 4 + half * 2;
                pf[mt][0] = ldsP[(mt * 16 + lr) * 16 + col];
                pf[mt][1] = ldsP[(mt * 16 + lr) * 16 + col + 1];
            }
#pragma unroll
            for (int nt = 0; nt < 4; ++nt) {
                v2f vf;
                const int krow = kb + kk * 4 + half * 2;
                vf[0] = Vb[(size_t)(krow + 0) * H_ + nt * 16 + lr];
                vf[1] = Vb[(size_t)(krow + 1) * H_ + nt * 16 + lr];
                ctx[0][nt] = wmma_f32(pf[0], vf, ctx[0][nt]);
                ctx[1][nt] = wmma_f32(pf[1], vf, ctx[1][nt]);
            }
        }
        __syncthreads();
    }

#pragma unroll
    for (int mt = 0; mt < 2; ++mt)
#pragma unroll
        for (int nt = 0; nt < 4; ++nt)
#pragma unroll
            for (int r = 0; r < 8; ++r) {
                const int q = q0 + mt * 16 + half * 8 + r;
                Ctx[((size_t)b * LQ_ + q) * H_ + h * HD_ + nt * 16 + lr] =
                    ctx[mt][nt][r] / lrow[mt][r];
            }
}

__global__ __launch_bounds__(256)
void layernorm_kernel(const float* __restrict__ X, const float* __restrict__ g,
                      const float* __restrict__ beta, float* __restrict__ out)
{
    __shared__ float ssum[256];
    __shared__ float ssq[256];
    const int row = blockIdx.x;
    const float* __restrict__ x = X + (size_t)row * H_;
    float s = 0.f, s2 = 0.f;
    for (int i = threadIdx.x; i < H_; i += 256) { const float v = x[i]; s += v; s2 += v * v; }
    ssum[threadIdx.x] = s; ssq[threadIdx.x] = s2;
    __syncthreads();
    for (int off = 128; off > 0; off >>= 1) {
        if (threadIdx.x < off) {
            ssum[threadIdx.x] += ssum[threadIdx.x + off];
            ssq[threadIdx.x]  += ssq[threadIdx.x + off];
        }
        __syncthreads();
    }
    const float mean = ssum[0] * (1.0f / H_);
    const float var  = ssq[0] * (1.0f / H_) - mean * mean;
    const float rstd = rsqrtf(var + 1e-5f);
    for (int i = threadIdx.x; i < H_; i += 256)
        out[(size_t)row * H_ + i] = (x[i] - mean) * rstd * g[i] + beta[i];
}

extern "C" void kernel_launch(void* const* d_in, const int* in_sizes, int n_in,
                              void* d_out, int out_size, void* d_ws, size_t ws_size,
                              hipStream_t stream) {
    const float* query = (const float*)d_in[0];
    const float* key   = (const float*)d_in[1];
    const float* value = (const float*)d_in[2];
    const int*   mask  = (const int*)  d_in[3];
    const float* Wq = (const float*)d_in[4];
    const float* bq = (const float*)d_in[5];
    const float* Wk = (const float*)d_in[6];
    const float* bk = (const float*)d_in[7];
    const float* Wv = (const float*)d_in[8];
    const float* bv = (const float*)d_in[9];
    const float* Wo = (const float*)d_in[10];
    const float* bo = (const float*)d_in[11];
    const float* ln_g = (const float*)d_in[12];
    const float* ln_b = (const float*)d_in[13];
    float* out = (float*)d_out;

    const size_t MQ = (size_t)B_ * LQ_;   // 2048
    const size_t MK = (size_t)B_ * LK_;   // 4096
    float* ws  = (float*)d_ws;
    float* Qp  = ws;
    float* Kp  = Qp + MQ * H_;
    float* Vp  = Kp + MK * H_;
    float* Ctx = Vp + MK * H_;
    float* Xr  = Ctx + MQ * H_;

    const dim3 blk(128);
    gemm_nt_bias<<<dim3(MQ / 64, H_ / 64), blk, 0, stream>>>(query, Wq, bq, nullptr, Qp, (int)MQ, H_, H_);
    gemm_nt_bias<<<dim3(MK / 64, H_ / 64), blk, 0, stream>>>(key,   Wk, bk, nullptr, Kp, (int)MK, H_, H_);
    gemm_nt_bias<<<dim3(MK / 64, H_ / 64), blk, 0, stream>>>(value, Wv, bv, nullptr, Vp, (int)MK, H_, H_);

    attn_head<<<B_ * NH_ * (LQ_ / 32), 32, 0, stream>>>(Qp, Kp, Vp, mask, Ctx);

    gemm_nt_bias<<<dim3(MQ / 64, H_ / 64), blk, 0, stream>>>(Ctx, Wo, bo, query, Xr, (int)MQ, H_, H_);
    layernorm_kernel<<<(int)MQ, 256, 0, stream>>>(Xr, ln_g, ln_b, out);
}